// LSTMDetachLayer_730144441071
// MI455X (gfx1250) — compile-verified
//
#include <hip/hip_runtime.h>

// LSTM forward, T=256, B=64, IN=H=1024 (detach_mask is gradient-only -> ignored).
//   Phase 1: Gpre[t*B+b, 4H] = x[t,b,:] @ w_ih^T + (b_ih + b_hh)   (one big GEMM)
//   Phase 2: ONE persistent kernel, 64 blocks x 512 threads, loops t=0..255.
//     - Each block's w_hh slice (16 cols x 4 gates x 1024 K = 256 KB) is staged
//       into LDS ONCE (padded rows: conflict-free b64 reads), then reused for
//       all 256 steps -> w_hh never touches L2 again on the critical path.
//     - gates = Gpre + h_prev @ w_hh^T with K split 4-way across waves,
//       partials reduced through LDS, fused pointwise LSTM update.
//     - Device-scope atomic barrier between steps (64 co-resident blocks).
// All matmuls via V_WMMA_F32_16X16X4_F32 (exact fp32).

#define T_STEPS 256
#define BATCH   64
#define IN_DIM  1024
#define H_DIM   1024
#define G4      (4 * H_DIM)   // 4096
#define NBLOCKS 64            // persistent blocks (one per 16 h-columns)

#define W_STRIDE 1028         // padded LDS row stride (dwords): bank offset 4*(rg&15)
#define W_LDS_FLOATS (64 * W_STRIDE)            // 65,792 floats = 263,168 B
#define RED_FLOATS   (12 * 1024)                // 48 KB reduction buffer
#define SMEM_BYTES   ((W_LDS_FLOATS + RED_FLOATS) * sizeof(float))  // 312,320 B

typedef __attribute__((ext_vector_type(2))) float v2f;
typedef __attribute__((ext_vector_type(4))) float v4f;
typedef __attribute__((ext_vector_type(8))) float v8f;

__device__ __forceinline__ v8f wmma_f32x4(v2f a, v2f b, v8f c) {
  // (neg_a, A, neg_b, B, c_mod, C, reuse_a, reuse_b)
  return __builtin_amdgcn_wmma_f32_16x16x4_f32(false, a, false, b, (short)0, c,
                                               false, false);
}

__device__ __forceinline__ float sigmoid_f(float x) {
  return 1.0f / (1.0f + __expf(-x));
}

// ---------------------------------------------------------------------------
// Phase 1: Gpre = X @ w_ih^T + (b_ih + b_hh)
// grid = (4H/16, (T*B)/(16*8)), block = 256 (8 waves, one 16x16 tile per wave)
// ---------------------------------------------------------------------------
__global__ __launch_bounds__(256) void lstm_pre_kernel(
    const float* __restrict__ x, const float* __restrict__ w_ih,
    const float* __restrict__ b_ih, const float* __restrict__ b_hh,
    float* __restrict__ gpre) {
  const int lane = threadIdx.x & 31;
  const int wave = threadIdx.x >> 5;
  const int half = lane >> 4;   // 0: lanes 0-15, 1: lanes 16-31
  const int l16  = lane & 15;

  const int mt = blockIdx.y * 8 + wave;  // 0..1023  (M tile over T*B)
  const int nt = blockIdx.x;             // 0..255   (N tile over 4H)

  const int mA = mt * 16 + l16;
  const int n  = nt * 16 + l16;

  const float* __restrict__ arow = x    + (size_t)mA * IN_DIM;
  const float* __restrict__ brow = w_ih + (size_t)n  * IN_DIM;

  v8f acc = {};
#pragma unroll 4
  for (int k0 = 0; k0 < IN_DIM; k0 += 4) {
    const int kk = k0 + half * 2;  // lanes 0-15: K={k0,k0+1}; lanes 16-31: K={k0+2,k0+3}
    v2f a = *(const v2f*)(arow + kk);
    v2f b = *(const v2f*)(brow + kk);
    acc = wmma_f32x4(a, b, acc);
  }

  const float bias = b_ih[n] + b_hh[n];
#pragma unroll
  for (int r = 0; r < 8; ++r) {
    const int m = mt * 16 + half * 8 + r;   // C/D layout: VGPR r -> M=r / M=8+r
    gpre[(size_t)m * G4 + n] = acc[r] + bias;
  }
}

// ---------------------------------------------------------------------------
// Barrier-counter init (replay determinism: counter must restart at 0).
// ---------------------------------------------------------------------------
__global__ void bar_init_kernel(unsigned* bar) { *bar = 0u; }

// ---------------------------------------------------------------------------
// Phase 2: persistent recurrent kernel.
// 64 blocks x 512 threads (16 waves). Wave w: m_tile = w&3, kslice = w>>2.
// w_hh B-operands come from LDS (staged once); A (=h) comes from L2.
// ---------------------------------------------------------------------------
__global__ __launch_bounds__(512) void lstm_seq_kernel(
    const float* __restrict__ h0,      // [B, H]
    const float* __restrict__ c0,      // [B, H]
    const float* __restrict__ w_hh,    // [4H, H]
    const float* __restrict__ gpre,    // [T*B, 4H] or nullptr
    const float* __restrict__ x,       // [T, B, IN]  (fused fallback)
    const float* __restrict__ w_ih,    // [4H, IN]    (fused fallback)
    const float* __restrict__ b_ih, const float* __restrict__ b_hh,
    float* __restrict__ out,           // [T, B, H]
    float* __restrict__ cbuf,          // [B, H] live c state (cT output slot)
    unsigned* __restrict__ bar) {
  extern __shared__ float smem[];
  float* __restrict__ wlds = smem;                 // [64][W_STRIDE] w_hh slice
  float* __restrict__ red  = smem + W_LDS_FLOATS;  // [12][1024] K-partials

  const int lane   = threadIdx.x & 31;
  const int wave   = threadIdx.x >> 5;
  const int half   = lane >> 4;
  const int l16    = lane & 15;
  const int mtile  = wave & 3;          // batch tile 0..3
  const int kslice = wave >> 2;         // K slice 0..3

  const int mbase = mtile * 16;
  const int mA    = mbase + l16;                // A row for this lane
  const int nbase = blockIdx.x * 16;
  const int ncol  = nbase + l16;                // h column for this lane
  const int kbeg  = kslice * (H_DIM / 4);       // 256-wide K slice

  // ---- stage this block's w_hh slice into LDS (once) ----
  // logical row rg = g*16 + c  ->  w_hh row g*H + nbase + c, padded stride 1028.
  for (int e = threadIdx.x; e < 64 * 256; e += 512) {
    const int rg = e >> 8;          // 0..63
    const int k4 = (e & 255) * 4;   // 0..1020
    const int g  = rg >> 4;
    const int c  = rg & 15;
    v4f v = *(const v4f*)(w_hh + (size_t)(g * H_DIM + nbase + c) * H_DIM + k4);
    *(v4f*)(wlds + rg * W_STRIDE + k4) = v;
  }
  __syncthreads();

  const float* __restrict__ lw0 = wlds + (0 * 16 + l16) * W_STRIDE + kbeg;
  const float* __restrict__ lw1 = wlds + (1 * 16 + l16) * W_STRIDE + kbeg;
  const float* __restrict__ lw2 = wlds + (2 * 16 + l16) * W_STRIDE + kbeg;
  const float* __restrict__ lw3 = wlds + (3 * 16 + l16) * W_STRIDE + kbeg;

  // Fused-fallback weight rows (x @ w_ih^T computed here if no Gpre scratch).
  const float* __restrict__ u0 = w_ih + (size_t)(0 * H_DIM + ncol) * IN_DIM + kbeg;
  const float* __restrict__ u1 = w_ih + (size_t)(1 * H_DIM + ncol) * IN_DIM + kbeg;
  const float* __restrict__ u2 = w_ih + (size_t)(2 * H_DIM + ncol) * IN_DIM + kbeg;
  const float* __restrict__ u3 = w_ih + (size_t)(3 * H_DIM + ncol) * IN_DIM + kbeg;

  const float bias_i = (gpre == nullptr) ? (b_ih[0 * H_DIM + ncol] + b_hh[0 * H_DIM + ncol]) : 0.0f;
  const float bias_f = (gpre == nullptr) ? (b_ih[1 * H_DIM + ncol] + b_hh[1 * H_DIM + ncol]) : 0.0f;
  const float bias_g = (gpre == nullptr) ? (b_ih[2 * H_DIM + ncol] + b_hh[2 * H_DIM + ncol]) : 0.0f;
  const float bias_o = (gpre == nullptr) ? (b_ih[3 * H_DIM + ncol] + b_hh[3 * H_DIM + ncol]) : 0.0f;

  for (int t = 0; t < T_STEPS; ++t) {
    const float* __restrict__ hprev =
        (t == 0) ? h0 : out + (size_t)(t - 1) * BATCH * H_DIM;
    const float* __restrict__ arow = hprev + (size_t)mA * H_DIM + kbeg;

    v8f acc0 = {}, acc1 = {}, acc2 = {}, acc3 = {};
#pragma unroll 4
    for (int k0 = 0; k0 < H_DIM / 4; k0 += 4) {
      const int kk = k0 + half * 2;
      v2f a  = *(const v2f*)(arow + kk);   // global (L2): h from other blocks
      v2f w0 = *(const v2f*)(lw0 + kk);    // LDS, conflict-free
      v2f w1 = *(const v2f*)(lw1 + kk);
      v2f w2 = *(const v2f*)(lw2 + kk);
      v2f w3 = *(const v2f*)(lw3 + kk);
      acc0 = wmma_f32x4(a, w0, acc0);
      acc1 = wmma_f32x4(a, w1, acc1);
      acc2 = wmma_f32x4(a, w2, acc2);
      acc3 = wmma_f32x4(a, w3, acc3);
    }

    if (gpre == nullptr) {  // uniform branch, EXEC stays full around WMMA
      const float* __restrict__ xrow =
          x + (size_t)t * BATCH * IN_DIM + (size_t)mA * IN_DIM + kbeg;
#pragma unroll 4
      for (int k0 = 0; k0 < IN_DIM / 4; k0 += 4) {
        const int kk = k0 + half * 2;
        v2f a  = *(const v2f*)(xrow + kk);
        v2f w0 = *(const v2f*)(u0 + kk);
        v2f w1 = *(const v2f*)(u1 + kk);
        v2f w2 = *(const v2f*)(u2 + kk);
        v2f w3 = *(const v2f*)(u3 + kk);
        acc0 = wmma_f32x4(a, w0, acc0);
        acc1 = wmma_f32x4(a, w1, acc1);
        acc2 = wmma_f32x4(a, w2, acc2);
        acc3 = wmma_f32x4(a, w3, acc3);
      }
    }

    // --- reduce K slices through LDS (layout [j*32+lane]: conflict-free) ---
    if (kslice > 0) {
      float* dst = &red[((kslice - 1) * 4 + mtile) * 1024];
#pragma unroll
      for (int r = 0; r < 8; ++r) {
        dst[(0 * 8 + r) * 32 + lane] = acc0[r];
        dst[(1 * 8 + r) * 32 + lane] = acc1[r];
        dst[(2 * 8 + r) * 32 + lane] = acc2[r];
        dst[(3 * 8 + r) * 32 + lane] = acc3[r];
      }
    }
    __syncthreads();

    if (kslice == 0) {
#pragma unroll
      for (int s = 0; s < 3; ++s) {
        const float* src = &red[(s * 4 + mtile) * 1024];
#pragma unroll
        for (int r = 0; r < 8; ++r) {
          acc0[r] += src[(0 * 8 + r) * 32 + lane];
          acc1[r] += src[(1 * 8 + r) * 32 + lane];
          acc2[r] += src[(2 * 8 + r) * 32 + lane];
          acc3[r] += src[(3 * 8 + r) * 32 + lane];
        }
      }

      const float* gp =
          (gpre != nullptr) ? gpre + (size_t)t * BATCH * G4 : nullptr;
      float* __restrict__ hout = out + (size_t)t * BATCH * H_DIM;
#pragma unroll
      for (int r = 0; r < 8; ++r) {
        const int m = mbase + half * 8 + r;
        float gi = acc0[r] + bias_i;
        float gf = acc1[r] + bias_f;
        float gg = acc2[r] + bias_g;
        float go = acc3[r] + bias_o;
        if (gp != nullptr) {
          const float* gpr = gp + (size_t)m * G4;
          gi += gpr[0 * H_DIM + ncol];
          gf += gpr[1 * H_DIM + ncol];
          gg += gpr[2 * H_DIM + ncol];
          go += gpr[3 * H_DIM + ncol];
        }
        const float ig = sigmoid_f(gi);
        const float fg = sigmoid_f(gf);
        const float gt = tanhf(gg);
        const float og = sigmoid_f(go);
        const float c2 = fg * cbuf[(size_t)m * H_DIM + ncol] + ig * gt;
        const float h2 = og * tanhf(c2);
        cbuf[(size_t)m * H_DIM + ncol] = c2;
        hout[(size_t)m * H_DIM + ncol] = h2;
      }
    }

    // --- device-scope barrier across the 64 persistent blocks ---
    __syncthreads();  // all waves done with LDS + this step's stores issued
    if (threadIdx.x == 0) {
      __threadfence();  // release our h/c stores to device scope
      __hip_atomic_fetch_add(bar, 1u, __ATOMIC_ACQ_REL, __HIP_MEMORY_SCOPE_AGENT);
      const unsigned target = (unsigned)NBLOCKS * (unsigned)(t + 1);
      while (__hip_atomic_load(bar, __ATOMIC_ACQUIRE, __HIP_MEMORY_SCOPE_AGENT) < target) {
        __builtin_amdgcn_s_sleep(2);
      }
    }
    __syncthreads();  // everyone proceeds with the new h visible
  }
}

extern "C" void kernel_launch(void* const* d_in, const int* in_sizes, int n_in,
                              void* d_out, int out_size, void* d_ws, size_t ws_size,
                              hipStream_t stream) {
  const float* x    = (const float*)d_in[0];  // [T, B, IN]
  const float* h0   = (const float*)d_in[1];  // [B, H]
  const float* c0   = (const float*)d_in[2];  // [B, H]
  const float* w_ih = (const float*)d_in[3];  // [4H, IN]
  const float* w_hh = (const float*)d_in[4];  // [4H, H]
  const float* b_ih = (const float*)d_in[5];  // [4H]
  const float* b_hh = (const float*)d_in[6];  // [4H]
  // d_in[7] = detach_mask: no effect on forward values.

  float* out  = (float*)d_out;                         // [T, B, H]
  float* hT   = out + (size_t)T_STEPS * BATCH * H_DIM; // [B, H]
  float* cbuf = hT + (size_t)BATCH * H_DIM;            // cT slot = live c state

  // Workspace layout: [barrier counter (256 B)] [Gpre 256 MB].
  unsigned* bar = (unsigned*)d_ws;
  const size_t gpre_bytes = (size_t)T_STEPS * BATCH * G4 * sizeof(float);
  float* gpre = (ws_size >= 256 + gpre_bytes) ? (float*)((char*)d_ws + 256) : nullptr;

  // Allow >64KB dynamic LDS for the persistent kernel (320 KB/WGP on CDNA5).
  hipFuncSetAttribute((const void*)lstm_seq_kernel,
                      hipFuncAttributeMaxDynamicSharedMemorySize,
                      (int)SMEM_BYTES);

  bar_init_kernel<<<1, 1, 0, stream>>>(bar);

  if (gpre != nullptr) {
    dim3 grid(G4 / 16, (T_STEPS * BATCH) / (16 * 8));
    lstm_pre_kernel<<<grid, 256, 0, stream>>>(x, w_ih, b_ih, b_hh, gpre);
  }

  // Initialize live c state from c0 (cbuf is in d_out, poisoned by harness).
  hipMemcpyAsync(cbuf, c0, (size_t)BATCH * H_DIM * sizeof(float),
                 hipMemcpyDeviceToDevice, stream);

  lstm_seq_kernel<<<dim3(NBLOCKS), 512, SMEM_BYTES, stream>>>(
      h0, c0, w_hh, gpre, x, w_ih, b_ih, b_hh, out, cbuf, bar);

  // hT = h at last step (cT already lives in its output slot).
  hipMemcpyAsync(hT, out + (size_t)(T_STEPS - 1) * BATCH * H_DIM,
                 (size_t)BATCH * H_DIM * sizeof(float),
                 hipMemcpyDeviceToDevice, stream);
}